// CanvasLayer_12000138625030
// MI455X (gfx1250) — compile-verified
//
#include <hip/hip_runtime.h>
#include <math.h>

// ---------------- problem constants (match reference) ----------------
#define B_N   4
#define K_N   1024
#define D_N   192
#define TOK   (B_N * K_N)          // 4096 tokens
#define HIMG  512
#define EPSLN 1e-5f

// canvas element counts
#define C2_ELEMS (B_N * D_N * 128 * 128)   // 12,582,912
#define C3_ELEMS (B_N * D_N * 64 * 64)     //  3,145,728
#define C4_ELEMS (B_N * D_N * 32 * 32)     //    786,432
#define H_ELEMS  (TOK * D_N)               //    786,432

typedef float v2f __attribute__((ext_vector_type(2)));
typedef float v8f __attribute__((ext_vector_type(8)));

__device__ __forceinline__ float sigm(float x) { return 1.f / (1.f + expf(-x)); }
__device__ __forceinline__ float gelu_exact(float x) {
  return 0.5f * x * (1.f + erff(x * 0.70710678118654752440f));
}

// ---------------- WMMA fp32 GEMM:  C[M,N] = A[M,K] * W[N,K]^T + bias[N] -----
// One 16x16 output tile per wave32; K stepped by 4 via V_WMMA_F32_16X16X4_F32.
// A-frag & B-frag share the same per-lane pattern: row = lane&15,
// k-pair = kbase + (lane>>4)*2  (ISA 7.12.2, 32-bit A 16x4 layout).
__global__ void gemm_wmma_nt(const float* __restrict__ A,
                             const float* __restrict__ W,
                             const float* __restrict__ bias,
                             float* __restrict__ C,
                             int M, int N, int Kd) {
  const int lane  = threadIdx.x & 31;
  const int wave  = threadIdx.x >> 5;            // 4 waves / block
  const int tileM = blockIdx.x * 4 + wave;
  const int tileN = blockIdx.y;
  if ((tileM << 4) >= M) return;                 // wave-uniform: EXEC stays all-1s
  const int m0 = tileM << 4;
  const int n0 = tileN << 4;
  const int r  = lane & 15;
  const int kk = (lane >> 4) << 1;

  const float* Ar = A + (size_t)(m0 + r) * Kd + kk;
  const float* Wr = W + (size_t)(n0 + r) * Kd + kk;

  v8f acc = {};
  for (int k = 0; k < Kd; k += 4) {
    __builtin_prefetch(Ar + k + 128, 0, 0);      // -> global_prefetch_b8
    v2f a = *(const v2f*)(Ar + k);
    v2f b = *(const v2f*)(Wr + k);
    acc = __builtin_amdgcn_wmma_f32_16x16x4_f32(
        /*neg_a=*/false, a, /*neg_b=*/false, b,
        /*c_mod=*/(short)0, acc, /*reuse_a=*/false, /*reuse_b=*/false);
  }

  // C/D layout: vgpr i -> m = m0 + i + (lane>>4)*8 ; n = n0 + (lane&15)
  const int n  = n0 + (lane & 15);
  const int mb = m0 + ((lane >> 4) << 3);
  const float bv = bias[n];
#pragma unroll
  for (int i = 0; i < 8; ++i)
    C[(size_t)(mb + i) * N + n] = acc[i] + bv;
}

// ---------------- zero fill ----------------
__global__ void zero_k(float* __restrict__ p, int n) {
  int i = blockIdx.x * 256 + threadIdx.x;
  if (i < n) p[i] = 0.f;
}

// ---------------- scatter-add tokens into canvas ----------------
// proj: (B,K,D) ; coords: (B,K,2) ints in [0,512) ; canvas: (B,D,Hc,Wc)
__global__ void scatter_add_k(const float* __restrict__ proj,
                              const int* __restrict__ coords,
                              float* __restrict__ canvas,
                              int Hc, int Wc, int stride) {
  int t = blockIdx.x;                 // token
  int d = threadIdx.x;                // channel, blockDim == D_N
  int b = t >> 10;                    // t / K_N
  int cx = coords[t * 2 + 0] / stride;
  int cy = coords[t * 2 + 1] / stride;
  int pix = cy * Wc + cx;
  atomicAdd(canvas + (size_t)(b * D_N + d) * Hc * Wc + pix,
            proj[(size_t)t * D_N + d]);
}

// ---------------- bilinear upsample (align-corners linspace) ----------------
__global__ void bilinear_up_k(const float* __restrict__ src, float* __restrict__ dst,
                              int h, int w, int oh, int ow) {
  int idx = blockIdx.x * blockDim.x + threadIdx.x;
  int total = B_N * D_N * oh * ow;
  if (idx >= total) return;
  int ox = idx % ow;
  int oy = (idx / ow) % oh;
  int c  = idx / (ow * oh);           // b*D + d
  float ys = (float)oy * (float)(h - 1) / (float)(oh - 1);
  float xs = (float)ox * (float)(w - 1) / (float)(ow - 1);
  int y0 = (int)floorf(ys); int y1 = y0 + 1; if (y1 > h - 1) y1 = h - 1;
  int x0 = (int)floorf(xs); int x1 = x0 + 1; if (x1 > w - 1) x1 = w - 1;
  float wy = ys - (float)y0, wx = xs - (float)x0;
  const float* s = src + (size_t)c * h * w;
  float v00 = s[y0 * w + x0], v01 = s[y0 * w + x1];
  float v10 = s[y1 * w + x0], v11 = s[y1 * w + x1];
  dst[idx] = (v00 * (1.f - wx) + v01 * wx) * (1.f - wy)
           + (v10 * (1.f - wx) + v11 * wx) * wy;
}

// ---------------- gate = sigmoid(dw3x3(cf,wf)+bf + dw3x3(cu,wc)+bc) ----------
__global__ void gate_k(const float* __restrict__ cf, const float* __restrict__ cu,
                       const float* __restrict__ wf, const float* __restrict__ bf,
                       const float* __restrict__ wc, const float* __restrict__ bc,
                       float* __restrict__ gate, int Hc, int Wc) {
  int idx = blockIdx.x * blockDim.x + threadIdx.x;
  int total = B_N * D_N * Hc * Wc;
  if (idx >= total) return;
  int x = idx % Wc;
  int y = (idx / Wc) % Hc;
  int ch = idx / (Wc * Hc);
  int d = ch % D_N;
  const float* pf = cf + (size_t)ch * Hc * Wc;
  const float* pu = cu + (size_t)ch * Hc * Wc;
  const float* kf = wf + d * 9;
  const float* kc = wc + d * 9;
  float s = bf[d] + bc[d];
#pragma unroll
  for (int ky = 0; ky < 3; ++ky) {
    int yy = y + ky - 1;
    if (yy < 0 || yy >= Hc) continue;
#pragma unroll
    for (int kx = 0; kx < 3; ++kx) {
      int xx = x + kx - 1;
      if (xx < 0 || xx >= Wc) continue;
      float wA = kf[ky * 3 + kx], wB = kc[ky * 3 + kx];
      s += pf[yy * Wc + xx] * wA + pu[yy * Wc + xx] * wB;
    }
  }
  gate[idx] = sigm(s);
}

// ---------------- c += gate * up ----------------
__global__ void fuse_add_k(float* __restrict__ c, const float* __restrict__ gate,
                           const float* __restrict__ up, int n) {
  int i = blockIdx.x * 256 + threadIdx.x;
  if (i < n) c[i] += gate[i] * up[i];
}

// ---------------- tmp = gelu(dwconv(c,w)+b) ----------------
__global__ void dwconv_gelu_k(const float* __restrict__ c, const float* __restrict__ w,
                              const float* __restrict__ bias, float* __restrict__ out,
                              int Hc, int Wc, int ks) {
  int idx = blockIdx.x * blockDim.x + threadIdx.x;
  int total = B_N * D_N * Hc * Wc;
  if (idx >= total) return;
  int x = idx % Wc;
  int y = (idx / Wc) % Hc;
  int ch = idx / (Wc * Hc);
  int d = ch % D_N;
  int pad = ks >> 1;
  const float* p = c + (size_t)ch * Hc * Wc;
  const float* kd = w + d * ks * ks;
  float s = bias[d];
  for (int ky = 0; ky < ks; ++ky) {
    int yy = y + ky - pad;
    if (yy < 0 || yy >= Hc) continue;
    for (int kx = 0; kx < ks; ++kx) {
      int xx = x + kx - pad;
      if (xx < 0 || xx >= Wc) continue;
      s += p[yy * Wc + xx] * kd[ky * ks + kx];
    }
  }
  out[idx] = gelu_exact(s);
}

// ---------------- c += dwconv(tmp,w)+b  (writes own pixel only: race-free) ---
__global__ void dwconv_add_k(const float* __restrict__ t, const float* __restrict__ w,
                             const float* __restrict__ bias, float* __restrict__ c,
                             int Hc, int Wc, int ks) {
  int idx = blockIdx.x * blockDim.x + threadIdx.x;
  int total = B_N * D_N * Hc * Wc;
  if (idx >= total) return;
  int x = idx % Wc;
  int y = (idx / Wc) % Hc;
  int ch = idx / (Wc * Hc);
  int d = ch % D_N;
  int pad = ks >> 1;
  const float* p = t + (size_t)ch * Hc * Wc;
  const float* kd = w + d * ks * ks;
  float s = bias[d];
  for (int ky = 0; ky < ks; ++ky) {
    int yy = y + ky - pad;
    if (yy < 0 || yy >= Hc) continue;
    for (int kx = 0; kx < ks; ++kx) {
      int xx = x + kx - pad;
      if (xx < 0 || xx >= Wc) continue;
      s += p[yy * Wc + xx] * kd[ky * ks + kx];
    }
  }
  c[idx] += s;
}

// ---------------- per-token layernorm + scalar gate ----------------
__global__ void ln_gate_k(const float* __restrict__ h,
                          const float* __restrict__ g, const float* __restrict__ bb,
                          const float* __restrict__ Wg, const float* __restrict__ bg,
                          float* __restrict__ gate_tok) {
  __shared__ float sh[256];
  int t = blockIdx.x;
  int d = threadIdx.x;                // 256 threads, D_N=192 active
  float x = (d < D_N) ? h[(size_t)t * D_N + d] : 0.f;
  sh[d] = x; __syncthreads();
  for (int s = 128; s > 0; s >>= 1) { if (d < s) sh[d] += sh[d + s]; __syncthreads(); }
  float mean = sh[0] / (float)D_N; __syncthreads();
  float dx = (d < D_N) ? (x - mean) : 0.f;
  sh[d] = dx * dx; __syncthreads();
  for (int s = 128; s > 0; s >>= 1) { if (d < s) sh[d] += sh[d + s]; __syncthreads(); }
  float var = sh[0] / (float)D_N; __syncthreads();
  float contrib = 0.f;
  if (d < D_N)
    contrib = (dx * rsqrtf(var + EPSLN) * g[d] + bb[d]) * Wg[d];
  sh[d] = contrib; __syncthreads();
  for (int s = 128; s > 0; s >>= 1) { if (d < s) sh[d] += sh[d + s]; __syncthreads(); }
  if (d == 0) gate_tok[t] = sigm(sh[0] + bg[0]);
}

// ---------------- grid-sample gather -> rcat (TOK x 576) ----------------
__global__ void gather_rcat_k(const float* __restrict__ c2, const float* __restrict__ c3,
                              const float* __restrict__ c4, const float* __restrict__ grid,
                              float* __restrict__ rcat) {
  int t = blockIdx.x;                 // token
  int j = threadIdx.x;                // 0..575 (= 18 waves)
  int lvl = j / D_N;
  int d   = j % D_N;
  int b   = t >> 10;
  const float* c; int Hc, Wc;
  if (lvl == 0)      { c = c2; Hc = 128; Wc = 128; }
  else if (lvl == 1) { c = c3; Hc = 64;  Wc = 64;  }
  else               { c = c4; Hc = 32;  Wc = 32;  }
  float gx = grid[t * 2 + 0];
  float gy = grid[t * 2 + 1];
  float fx = (gx + 1.f) * 0.5f * (float)(Wc - 1);
  float fy = (gy + 1.f) * 0.5f * (float)(Hc - 1);
  float x0f = fminf(fmaxf(floorf(fx), 0.f), (float)(Wc - 1));
  float x1f = fminf(x0f + 1.f, (float)(Wc - 1));
  float y0f = fminf(fmaxf(floorf(fy), 0.f), (float)(Hc - 1));
  float y1f = fminf(y0f + 1.f, (float)(Hc - 1));
  float wx = fx - x0f, wy = fy - y0f;
  int x0 = (int)x0f, x1 = (int)x1f, y0 = (int)y0f, y1 = (int)y1f;
  const float* cd = c + (size_t)(b * D_N + d) * Hc * Wc;
  float v00 = cd[y0 * Wc + x0], v01 = cd[y0 * Wc + x1];
  float v10 = cd[y1 * Wc + x0], v11 = cd[y1 * Wc + x1];
  float top = v00 * (1.f - wx) + v01 * wx;
  float bot = v10 * (1.f - wx) + v11 * wx;
  rcat[(size_t)t * (3 * D_N) + lvl * D_N + d] = top * (1.f - wy) + bot * wy;
}

// ---------------- h_out = h + gate_tok * read ----------------
__global__ void combine_k(const float* __restrict__ h, const float* __restrict__ readv,
                          const float* __restrict__ gate_tok, float* __restrict__ out) {
  int i = blockIdx.x * 256 + threadIdx.x;
  if (i >= H_ELEMS) return;
  int t = i / D_N;
  out[i] = h[i] + gate_tok[t] * readv[i];
}

// =====================================================================
extern "C" void kernel_launch(void* const* d_in, const int* in_sizes, int n_in,
                              void* d_out, int out_size, void* d_ws, size_t ws_size,
                              hipStream_t stream) {
  // ---- inputs (setup_inputs dict order, params flattened in insertion order)
  const float* h           = (const float*)d_in[0];
  // d_in[1..3] = zero canvases (unused; we zero d_out regions ourselves)
  const float* center_grid = (const float*)d_in[4];   // (B,K,1,2)
  const int*   coords      = (const int*)d_in[5];     // (B,K,2)
  const float* Ww2   = (const float*)d_in[6];
  const float* bw2   = (const float*)d_in[7];
  const float* Ww3   = (const float*)d_in[8];
  const float* bw3   = (const float*)d_in[9];
  const float* Ww4   = (const float*)d_in[10];
  const float* bw4   = (const float*)d_in[11];
  const float* g43f_w = (const float*)d_in[12];
  const float* g43f_b = (const float*)d_in[13];
  const float* g43c_w = (const float*)d_in[14];
  const float* g43c_b = (const float*)d_in[15];
  const float* g32f_w = (const float*)d_in[16];
  const float* g32f_b = (const float*)d_in[17];
  const float* g32c_w = (const float*)d_in[18];
  const float* g32c_b = (const float*)d_in[19];
  const float* s2a_w = (const float*)d_in[20];
  const float* s2a_b = (const float*)d_in[21];
  const float* s2b_w = (const float*)d_in[22];
  const float* s2b_b = (const float*)d_in[23];
  const float* s3a_w = (const float*)d_in[24];
  const float* s3a_b = (const float*)d_in[25];
  const float* s3b_w = (const float*)d_in[26];
  const float* s3b_b = (const float*)d_in[27];
  const float* s4a_w = (const float*)d_in[28];
  const float* s4a_b = (const float*)d_in[29];
  const float* s4b_w = (const float*)d_in[30];
  const float* s4b_b = (const float*)d_in[31];
  const float* ln_g  = (const float*)d_in[32];
  const float* ln_b  = (const float*)d_in[33];
  const float* Wr    = (const float*)d_in[34];   // (D, 3D)
  const float* br    = (const float*)d_in[35];
  const float* Wg    = (const float*)d_in[36];   // (1, D)
  const float* bg    = (const float*)d_in[37];

  // ---- outputs (concatenated flat, return order)
  float* out_h = (float*)d_out;
  float* c2    = out_h + H_ELEMS;
  float* c3    = c2 + C2_ELEMS;
  float* c4    = c3 + C3_ELEMS;

  // ---- workspace aliasing (two 50MB regions, sequential lifetime)
  float* bufA = (float*)d_ws;               // C2_ELEMS floats
  float* bufB = bufA + C2_ELEMS;            // C2_ELEMS floats
  float* proj2   = bufA;                    // TOK*D each, dead before l3up
  float* proj3   = bufA + H_ELEMS;
  float* proj4   = bufA + 2 * H_ELEMS;
  float* l4up    = bufB;                    // C3_ELEMS, dead after fuse43
  float* gate43  = bufB + C3_ELEMS;         // C3_ELEMS, dead after fuse43
  float* l3up    = bufA;                    // C2_ELEMS
  float* gate32  = bufB;                    // C2_ELEMS, dead after fuse32
  float* tmp     = bufA;                    // smooth scratch (max C2_ELEMS)
  float* rcat    = bufA;                    // TOK*576, after smooths
  float* readv   = bufB;                    // TOK*D, after gate32 dead
  float* gate_tok = bufB + H_ELEMS;         // TOK floats

  const int EB = 256;
  // 1) zero canvases (c2,c3,c4 contiguous in d_out)
  {
    int n = C2_ELEMS + C3_ELEMS + C4_ELEMS;
    zero_k<<<(n + EB - 1) / EB, EB, 0, stream>>>(c2, n);
  }

  // 2) projection GEMMs via WMMA f32 16x16x4: (4096x192) = h(4096x192)*W^T
  {
    dim3 grid(TOK / 64, D_N / 16);          // 4 tiles(M) per 128-thread block
    gemm_wmma_nt<<<grid, 128, 0, stream>>>(h, Ww2, bw2, proj2, TOK, D_N, D_N);
    gemm_wmma_nt<<<grid, 128, 0, stream>>>(h, Ww3, bw3, proj3, TOK, D_N, D_N);
    gemm_wmma_nt<<<grid, 128, 0, stream>>>(h, Ww4, bw4, proj4, TOK, D_N, D_N);
  }

  // 3) scatter-add into canvases
  scatter_add_k<<<TOK, D_N, 0, stream>>>(proj2, coords, c2, 128, 128, 4);
  scatter_add_k<<<TOK, D_N, 0, stream>>>(proj3, coords, c3, 64, 64, 8);
  scatter_add_k<<<TOK, D_N, 0, stream>>>(proj4, coords, c4, 32, 32, 16);

  // 4) L4 -> L3 gated fusion
  bilinear_up_k<<<(C3_ELEMS + EB - 1) / EB, EB, 0, stream>>>(c4, l4up, 32, 32, 64, 64);
  gate_k<<<(C3_ELEMS + EB - 1) / EB, EB, 0, stream>>>(
      c3, l4up, g43f_w, g43f_b, g43c_w, g43c_b, gate43, 64, 64);
  fuse_add_k<<<(C3_ELEMS + EB - 1) / EB, EB, 0, stream>>>(c3, gate43, l4up, C3_ELEMS);

  // 5) L3 -> L2 gated fusion
  bilinear_up_k<<<(C2_ELEMS + EB - 1) / EB, EB, 0, stream>>>(c3, l3up, 64, 64, 128, 128);
  gate_k<<<(C2_ELEMS + EB - 1) / EB, EB, 0, stream>>>(
      c2, l3up, g32f_w, g32f_b, g32c_w, g32c_b, gate32, 128, 128);
  fuse_add_k<<<(C2_ELEMS + EB - 1) / EB, EB, 0, stream>>>(c2, gate32, l3up, C2_ELEMS);

  // 6) smooth each level: c += dwconv(gelu(dwconv(c)))
  dwconv_gelu_k<<<(C2_ELEMS + EB - 1) / EB, EB, 0, stream>>>(c2, s2a_w, s2a_b, tmp, 128, 128, 5);
  dwconv_add_k <<<(C2_ELEMS + EB - 1) / EB, EB, 0, stream>>>(tmp, s2b_w, s2b_b, c2, 128, 128, 5);
  dwconv_gelu_k<<<(C3_ELEMS + EB - 1) / EB, EB, 0, stream>>>(c3, s3a_w, s3a_b, tmp, 64, 64, 5);
  dwconv_add_k <<<(C3_ELEMS + EB - 1) / EB, EB, 0, stream>>>(tmp, s3b_w, s3b_b, c3, 64, 64, 5);
  dwconv_gelu_k<<<(C4_ELEMS + EB - 1) / EB, EB, 0, stream>>>(c4, s4a_w, s4a_b, tmp, 32, 32, 3);
  dwconv_add_k <<<(C4_ELEMS + EB - 1) / EB, EB, 0, stream>>>(tmp, s4b_w, s4b_b, c4, 32, 32, 3);

  // 7) layernorm + scalar gate per token
  ln_gate_k<<<TOK, 256, 0, stream>>>(h, ln_g, ln_b, Wg, bg, gate_tok);

  // 8) grid-sample reads -> rcat (TOK x 576)
  gather_rcat_k<<<TOK, 3 * D_N, 0, stream>>>(c2, c3, c4, center_grid, rcat);

  // 9) read projection GEMM: readv(4096x192) = rcat(4096x576) * Wr^T + br
  {
    dim3 grid(TOK / 64, D_N / 16);
    gemm_wmma_nt<<<grid, 128, 0, stream>>>(rcat, Wr, br, readv, TOK, D_N, 3 * D_N);
  }

  // 10) h_out = h + gate * read
  combine_k<<<(H_ELEMS + EB - 1) / EB, EB, 0, stream>>>(h, readv, gate_tok, out_h);
}